// GraphConvBlock_82051055222845
// MI455X (gfx1250) — compile-verified
//
#include <hip/hip_runtime.h>

#define N_NODES 100000
#define N_EDGES 1600000
#define DIM 64
#define LN_EPS 1e-5f

typedef float v2f __attribute__((ext_vector_type(2)));
typedef float v8f __attribute__((ext_vector_type(8)));

// ---------------------------------------------------------------------------
// Utility: zero a buffer (float4 granularity)
// ---------------------------------------------------------------------------
__global__ __launch_bounds__(256) void zero_f32_kernel(float* __restrict__ p, int n4) {
    int i = blockIdx.x * blockDim.x + threadIdx.x;
    if (i < n4) {
        ((float4*)p)[i] = make_float4(0.f, 0.f, 0.f, 0.f);
    }
}

// ---------------------------------------------------------------------------
// Degree: deg[v] += 1 for every endpoint of every edge
// ---------------------------------------------------------------------------
__global__ __launch_bounds__(256) void degree_kernel(const long long* __restrict__ edges,
                                                     float* __restrict__ deg) {
    int i = blockIdx.x * blockDim.x + threadIdx.x;
    if (i >= N_EDGES) return;
    int s = (int)edges[2ll * i + 0];
    int d = (int)edges[2ll * i + 1];
    atomicAdd(deg + s, 1.0f);
    atomicAdd(deg + d, 1.0f);
}

// ---------------------------------------------------------------------------
// Scatter: agg[dst] += msg[src] and agg[src] += msg[dst] (bidirectional).
// 4 threads per directed edge; each thread moves 16 consecutive floats.
// msg/agg are L2-resident (25.6MB each << 192MB L2) so the random gathers
// and float atomics are served at L2 bandwidth.
// ---------------------------------------------------------------------------
__global__ __launch_bounds__(256) void scatter_kernel(const long long* __restrict__ edges,
                                                      const float* __restrict__ msg,
                                                      float* __restrict__ agg) {
    int tid  = blockIdx.x * blockDim.x + threadIdx.x;
    int de   = tid >> 2;              // directed edge index in [0, 2E)
    if (de >= 2 * N_EDGES) return;
    int part = (tid & 3) * 16;        // which 16-float chunk of the row
    int e    = (de < N_EDGES) ? de : de - N_EDGES;
    int a    = (int)edges[2ll * e + 0];   // src
    int b    = (int)edges[2ll * e + 1];   // dst
    int from, to;
    if (de < N_EDGES) { from = a; to = b; }   // msg[src] -> agg[dst]
    else              { from = b; to = a; }   // msg[dst] -> agg[src]

    const float4* m = (const float4*)(msg + (long long)from * DIM + part);
    float*        o = agg + (long long)to * DIM + part;
#pragma unroll
    for (int i = 0; i < 4; ++i) {
        float4 v = m[i];
        atomicAdd(o + 4 * i + 0, v.x);
        atomicAdd(o + 4 * i + 1, v.y);
        atomicAdd(o + 4 * i + 2, v.z);
        atomicAdd(o + 4 * i + 3, v.w);
    }
}

// ---------------------------------------------------------------------------
// WMMA core: one wave computes a 16x64 output tile of h @ W using
// V_WMMA_F32_16X16X4_F32 (exact fp32 path; 16 K-steps x 4 col-tiles).
// A layout (16x4 f32): lanes 0-15 hold K=0,1 ; lanes 16-31 hold K=2,3.
// B layout (4x16 f32): VGPR v holds rows K=v (lanes 0-15) and K=v+2 (16-31).
// C/D (16x16 f32): VGPR v = row M=v (lanes 0-15) / M=v+8 (lanes 16-31).
// ---------------------------------------------------------------------------
__device__ __forceinline__ void wmma_row_tile(const float* __restrict__ hrow,
                                              const float* __restrict__ W,
                                              int lane, v8f acc[4]) {
    const int ncol  = lane & 15;
    const int khalf = lane >> 4;
#pragma unroll
    for (int kk = 0; kk < 16; ++kk) {
        const int k0 = kk * 4 + khalf * 2;
        float2 av = *(const float2*)(hrow + k0);
        v2f a; a.x = av.x; a.y = av.y;
        const float* w0 = W + (long long)k0 * DIM + ncol;
        const float* w1 = w0 + DIM;
#pragma unroll
        for (int j = 0; j < 4; ++j) {
            v2f b; b.x = w0[16 * j]; b.y = w1[16 * j];
            acc[j] = __builtin_amdgcn_wmma_f32_16x16x4_f32(
                false, a, false, b, (short)0, acc[j], false, false);
        }
    }
}

// ---------------------------------------------------------------------------
// msg = h @ W1 + b1
// ---------------------------------------------------------------------------
__global__ __launch_bounds__(256) void gemm_msg_kernel(const float* __restrict__ h,
                                                       const float* __restrict__ W,
                                                       const float* __restrict__ bias,
                                                       float* __restrict__ out) {
    const int wave = blockIdx.x * (blockDim.x >> 5) + (threadIdx.x >> 5);
    const int lane = threadIdx.x & 31;
    if (wave >= N_NODES / 16) return;   // wave-uniform guard: EXEC stays all-1s

    v8f acc[4] = {};
    const float* hrow = h + ((long long)wave * 16 + (lane & 15)) * DIM;
    wmma_row_tile(hrow, W, lane, acc);

    const int ncol = lane & 15;
    const int half = lane >> 4;
#pragma unroll
    for (int v = 0; v < 8; ++v) {
        const int m = v + half * 8;
        float* orow = out + ((long long)wave * 16 + m) * DIM;
#pragma unroll
        for (int j = 0; j < 4; ++j) {
            orow[j * 16 + ncol] = acc[j][v] + bias[j * 16 + ncol];
        }
    }
}

// ---------------------------------------------------------------------------
// out = h @ W0 + b0 + agg/deg  [+ LayerNorm + ReLU if do_ln]
// One wave owns 16 rows; a full 64-wide row lives in one wave-half across
// 16 lanes x 4 accumulators, so LN reductions are shfl_xor over 1/2/4/8.
// ---------------------------------------------------------------------------
__global__ __launch_bounds__(256) void gemm_out_kernel(const float* __restrict__ h,
                                                       const float* __restrict__ W,
                                                       const float* __restrict__ bias,
                                                       const float* __restrict__ agg,
                                                       const float* __restrict__ deg,
                                                       const float* __restrict__ gamma,
                                                       const float* __restrict__ beta,
                                                       float* __restrict__ out,
                                                       int do_ln) {
    const int wave = blockIdx.x * (blockDim.x >> 5) + (threadIdx.x >> 5);
    const int lane = threadIdx.x & 31;
    if (wave >= N_NODES / 16) return;

    v8f acc[4] = {};
    const float* hrow = h + ((long long)wave * 16 + (lane & 15)) * DIM;
    wmma_row_tile(hrow, W, lane, acc);

    const int ncol = lane & 15;
    const int half = lane >> 4;
#pragma unroll
    for (int v = 0; v < 8; ++v) {
        const int m = v + half * 8;
        const long long row = (long long)wave * 16 + m;
        const float rdeg = 1.0f / fmaxf(deg[row], 1.0f);
        const float* arow = agg + row * DIM;

        float x[4];
        float s = 0.0f, q = 0.0f;
#pragma unroll
        for (int j = 0; j < 4; ++j) {
            x[j] = acc[j][v] + bias[j * 16 + ncol] + arow[j * 16 + ncol] * rdeg;
            s += x[j];
            q += x[j] * x[j];
        }

        float* orow = out + row * DIM;
        if (do_ln) {
            // reduce across the 16 lanes of this wave-half (offsets < 16
            // keep the xor inside the half)
#pragma unroll
            for (int off = 1; off < 16; off <<= 1) {
                s += __shfl_xor(s, off, 32);
                q += __shfl_xor(q, off, 32);
            }
            const float mean = s * (1.0f / DIM);
            const float var  = q * (1.0f / DIM) - mean * mean;
            const float rstd = rsqrtf(var + LN_EPS);
#pragma unroll
            for (int j = 0; j < 4; ++j) {
                float y = (x[j] - mean) * rstd * gamma[j * 16 + ncol] + beta[j * 16 + ncol];
                orow[j * 16 + ncol] = fmaxf(y, 0.0f);
            }
        } else {
#pragma unroll
            for (int j = 0; j < 4; ++j) {
                orow[j * 16 + ncol] = x[j];
            }
        }
    }
}

// ---------------------------------------------------------------------------
// Host-side launch sequence
// ---------------------------------------------------------------------------
extern "C" void kernel_launch(void* const* d_in, const int* in_sizes, int n_in,
                              void* d_out, int out_size, void* d_ws, size_t ws_size,
                              hipStream_t stream) {
    const float*     vert  = (const float*)d_in[0];
    const long long* edges = (const long long*)d_in[1];   // int64 per reference
    const float*     W0    = (const float*)d_in[2];
    const float*     b0    = (const float*)d_in[3];
    const float*     W1    = (const float*)d_in[4];
    const float*     b1    = (const float*)d_in[5];
    const float*     gamma = (const float*)d_in[6];
    const float*     beta  = (const float*)d_in[7];
    float*           out   = (float*)d_out;

    char* ws = (char*)d_ws;
    const size_t featBytes = (size_t)N_NODES * DIM * sizeof(float);  // 25.6 MB
    float* msg  = (float*)ws;  ws += featBytes;
    float* agg  = (float*)ws;  ws += featBytes;
    float* htmp = (float*)ws;  ws += featBytes;
    float* deg  = (float*)ws;  // N_NODES floats

    const int feat4 = N_NODES * DIM / 4;   // 1.6M float4
    const int deg4  = N_NODES / 4;         // 25000 float4 (N divisible by 4)

    // degree (computed once)
    zero_f32_kernel<<<(deg4 + 255) / 256, 256, 0, stream>>>(deg, deg4);
    degree_kernel<<<(N_EDGES + 255) / 256, 256, 0, stream>>>(edges, deg);

    const int gemmBlocks    = (N_NODES / 16 + 7) / 8;          // 8 waves/block
    const int scatterBlocks = (2 * N_EDGES * 4) / 256;          // 50000

    for (int l = 0; l < 3; ++l) {
        const float* hin  = (l == 0) ? vert : htmp;
        float*       hout = (l == 2) ? out  : htmp;  // in-place for l==1 is
                                                     // safe: each wave reads
                                                     // its 16 rows before
                                                     // storing them
        gemm_msg_kernel<<<gemmBlocks, 256, 0, stream>>>(
            hin, W1 + (long long)l * DIM * DIM, b1 + l * DIM, msg);
        zero_f32_kernel<<<(feat4 + 255) / 256, 256, 0, stream>>>(agg, feat4);
        scatter_kernel<<<scatterBlocks, 256, 0, stream>>>(edges, msg, agg);

        const int lg = (l < 2) ? l : 0;  // gamma/beta only have 2 rows
        gemm_out_kernel<<<gemmBlocks, 256, 0, stream>>>(
            hin, W0 + (long long)l * DIM * DIM, b0 + l * DIM,
            agg, deg, gamma + lg * DIM, beta + lg * DIM, hout, (l < 2) ? 1 : 0);
    }
}